// SimpleGCN_57999238365688
// MI455X (gfx1250) — compile-verified
//
#include <hip/hip_runtime.h>

typedef __attribute__((ext_vector_type(2))) float v2f;
typedef __attribute__((ext_vector_type(8))) float v8f;

#define HIDDEN 128
#define CHUNK  256

// -----------------------------------------------------------------------------
// Kernel 1: support = X @ W^T + bias, via V_WMMA_F32_16X16X4_F32 (full fp32).
//   Block = 256 threads = 8 waves. Each wave computes one 16(M) x 16(N) tile:
//   M = 16 rows of X starting at blockIdx.x*16, N = 16 output cols = wave*16.
//   K = HIDDEN = 128 -> 32 chained WMMA ops (K-step 4).
//
//   fp32 16x4 A layout (ISA 7.12.2): lanes 0-15 hold M=lane, K={0,1} in v[0],v[1];
//   lanes 16-31 hold M=lane-16, K={2,3}. B (4x16) is symmetric: lane = N column,
//   half selects K pair. Both fragments are one 8-byte load per lane per step.
// -----------------------------------------------------------------------------
__global__ __launch_bounds__(256) void gcn_gemm_wmma(
    const float* __restrict__ X, const float* __restrict__ W,
    const float* __restrict__ bias, float* __restrict__ support)
{
    const int wave = threadIdx.x >> 5;
    const int lane = threadIdx.x & 31;
    const int half = lane >> 4;   // 0: K pair {0,1}; 1: K pair {2,3}
    const int idx  = lane & 15;   // M for A, N for B
    const int row0 = blockIdx.x * 16;
    const int h0   = wave * 16;

    const float* ap = X + (size_t)(row0 + idx) * HIDDEN + 2 * half;
    const float* bp = W + (size_t)(h0  + idx) * HIDDEN + 2 * half;

    // Fold bias into accumulator: column h = h0 + idx is fixed per lane.
    const float bv = bias[h0 + idx];
    v8f c;
    #pragma unroll
    for (int i = 0; i < 8; ++i) c[i] = bv;

    #pragma unroll
    for (int kb = 0; kb < HIDDEN / 4; ++kb) {
        v2f a = *(const v2f*)(ap + kb * 4);
        v2f b = *(const v2f*)(bp + kb * 4);
        // 8 args: (neg_a, A, neg_b, B, c_mod, C, reuse_a, reuse_b)
        c = __builtin_amdgcn_wmma_f32_16x16x4_f32(
                false, a, false, b, (short)0, c, false, false);
    }

    // C/D layout: VGPR v -> row M = v + 8*half, col N = idx.
    float* op = support + (size_t)row0 * HIDDEN + h0 + idx;
    #pragma unroll
    for (int v = 0; v < 8; ++v)
        op[(size_t)(v + 8 * half) * HIDDEN] = c[v];
}

// -----------------------------------------------------------------------------
// Kernel 2: out[r,:] += val * support[c,:] with segment-run accumulation.
//   Block = 128 threads, thread t = column t. Each block walks CHUNK sorted
//   nnz entries; rows/cols/vals indices are block-uniform -> scalar loads.
//   Register accumulation while the row stays constant; atomic flush on change
//   (avg run length = NNZ/NUM_ITEMS = 16 -> ~16x less atomic traffic).
// -----------------------------------------------------------------------------
__global__ __launch_bounds__(128) void gcn_spmm(
    const float* __restrict__ support, const int* __restrict__ rows,
    const int* __restrict__ cols, const float* __restrict__ vals,
    float* __restrict__ out, int nnz)
{
    const int t = threadIdx.x;
    long e0 = (long)blockIdx.x * CHUNK;
    long e1 = e0 + CHUNK;
    if (e1 > nnz) e1 = nnz;
    if (e0 >= e1) return;

    float acc = 0.0f;
    int cur = rows[e0];

    for (long e = e0; e < e1; ++e) {
        const int   r = rows[e];
        const int   c = cols[e];
        const float v = vals[e];

        // Warm the L2-resident gather a few entries ahead (global_prefetch_b8).
        if (e + 8 < e1)
            __builtin_prefetch(&support[(size_t)cols[e + 8] * HIDDEN + t], 0, 1);

        if (r != cur) {
            atomicAdd(&out[(size_t)cur * HIDDEN + t], acc);
            acc = 0.0f;
            cur = r;
        }
        acc = fmaf(v, support[(size_t)c * HIDDEN + t], acc);
    }
    atomicAdd(&out[(size_t)cur * HIDDEN + t], acc);
}

// -----------------------------------------------------------------------------
// Launch: zero output, GEMM into workspace, then SpMM scatter into output.
// Inputs (setup_inputs order): X, adj_rows, adj_cols, adj_vals, W, bias.
// -----------------------------------------------------------------------------
extern "C" void kernel_launch(void* const* d_in, const int* in_sizes, int n_in,
                              void* d_out, int out_size, void* d_ws, size_t ws_size,
                              hipStream_t stream) {
    const float* X    = (const float*)d_in[0];
    const int*   rows = (const int*)  d_in[1];
    const int*   cols = (const int*)  d_in[2];
    const float* vals = (const float*)d_in[3];
    const float* W    = (const float*)d_in[4];
    const float* bias = (const float*)d_in[5];
    float*       out  = (float*)d_out;

    const int N   = in_sizes[0] / HIDDEN;  // 100000 (divisible by 16)
    const int nnz = in_sizes[1];           // 1600000

    float* support = (float*)d_ws;         // N*HIDDEN*4 = 51.2 MB scratch

    hipMemsetAsync(d_out, 0, (size_t)out_size * sizeof(float), stream);

    gcn_gemm_wmma<<<N / 16, 256, 0, stream>>>(X, W, bias, support);

    const int nblk = (nnz + CHUNK - 1) / CHUNK;
    gcn_spmm<<<nblk, 128, 0, stream>>>(support, rows, cols, vals, out, nnz);
}